// ASCADv2Head_66477503807523
// MI455X (gfx1250) — compile-verified
//
#include <hip/hip_runtime.h>

typedef __attribute__((ext_vector_type(2))) float v2f;
typedef __attribute__((ext_vector_type(8))) float v8f;

// ---------------- GF(256) tables at compile time (poly 0x11B, generator 3) ---
struct GF {
  unsigned short logt[256];
  unsigned short alog[256];
};

constexpr GF makeGF() {
  GF g{};
  int a = 1;
  for (int i = 0; i < 255; ++i) {
    g.alog[i] = (unsigned short)a;
    g.logt[a] = (unsigned short)i;
    int x = a << 1;
    if (x & 0x100) x ^= 0x11B;  // AES polynomial
    a = x ^ a;                  // a *= 3
  }
  g.logt[0] = 0;
  g.alog[255] = 1;
  return g;
}

__device__ __constant__ GF GFT = makeGF();

// ---------------------------------------------------------------------------
// One workgroup (256 threads = 8 wave32) per batch element b.
//   GEMM1: xorred[16,256] = msbox x B1,  B1[x,z] = beta[x^z]
//   GEMM2: out[16,256]    = xorred x B2, B2[y,z] = inv_alpha[DIV[y,z]]
// V_WMMA_F32_16X16X4_F32; each wave owns N-tiles {wave, wave+8}.
// GEMM1 B: byte-XOR addressing; half==1 lanes use a +16-dword-shifted beta
//          copy -> 32 lanes hit 32 distinct LDS banks (conflict-free).
// GEMM2: K-loop reordered by GF exponent (y = alog[254-j]) so B addresses are
//          SEQUENTIAL: B'[j,z] = G2[log z + j + 1]. Fully unrolled -> each B
//          pair is one ds_load_2addr_b32 with immediate offsets. A becomes the
//          scattered operand via a precomputed byte-offset table (uniform
//          ds_load_b64 per k-step). y=0 fixup row peeled into the LAST step.
//          z==0 lanes' base redirected into an invA0-filled G2 tail (no
//          selects in the loop).
// ---------------------------------------------------------------------------
__global__ __launch_bounds__(256)
void ascadv2_head_kernel(const float* __restrict__ logits,
                         float* __restrict__ out) {
  __shared__ __align__(16) float s_alpha[256];
  __shared__ __align__(16) float s_beta[256];
  __shared__ __align__(16) float s_beta2[272];      // beta shifted +16 dwords
  __shared__ __align__(16) float s_invA[256];
  __shared__ __align__(16) float s_msbox[16 * 256];
  __shared__ __align__(16) float s_xorT[256 * 17];  // xorred, [z][l], pad 17
  __shared__ __align__(16) float s_G2[768];         // [0,510): G wrapped 2x
                                                    // [510,768): invA0 fill
  __shared__ __align__(16) int   s_aoff[256];       // A byte-offsets, exp order
  __shared__ float s_sum;

  const int b    = blockIdx.x;
  const int t    = threadIdx.x;
  const int lane = t & 31;
  const int wave = t >> 5;
  const int m    = lane & 15;   // M row inside fragment / N column inside tile
  const int half = lane >> 4;   // 0: K pair {k0,k0+1}, 1: {k0+2,k0+3}

  const float* __restrict__ src = logits + (size_t)b * 18 * 256;

  // ---- Phase 1: softmax of each of 18 rows (rows striped over 8 waves) ----
  for (int r = wave; r < 18; r += 8) {
    const float4* row4 = (const float4*)(src + r * 256) + lane * 2;
    float4 u0 = row4[0];
    float4 u1 = row4[1];
    float v[8] = {u0.x, u0.y, u0.z, u0.w, u1.x, u1.y, u1.z, u1.w};
    float mx = -3.402823466e38f;
#pragma unroll
    for (int i = 0; i < 8; ++i) mx = fmaxf(mx, v[i]);
#pragma unroll
    for (int off = 16; off > 0; off >>= 1) mx = fmaxf(mx, __shfl_xor(mx, off, 32));
    float s = 0.f;
#pragma unroll
    for (int i = 0; i < 8; ++i) { v[i] = __expf(v[i] - mx); s += v[i]; }
#pragma unroll
    for (int off = 16; off > 0; off >>= 1) s += __shfl_xor(s, off, 32);
    const float rs = 1.0f / s;
    float* dst = (r == 0) ? s_alpha : (r == 1) ? s_beta : &s_msbox[(r - 2) * 256];
    float4 w0 = {v[0] * rs, v[1] * rs, v[2] * rs, v[3] * rs};
    float4 w1 = {v[4] * rs, v[5] * rs, v[6] * rs, v[7] * rs};
    ((float4*)dst)[lane * 2]     = w0;
    ((float4*)dst)[lane * 2 + 1] = w1;
  }
  // A-offset table in exponent order: j -> y = alog[254-j]; pad-17 T layout.
  s_aoff[t] = (t == 255) ? 0 : ((int)GFT.alog[254 - t]) * 68;
  __syncthreads();

  // ---- Phase 2: inv_alpha, shifted beta copy, G2 table, sum(inv_alpha) ----
  const int lt     = GFT.logt[t];
  const int invIdx = (t == 0) ? 0 : (int)GFT.alog[(255 - lt) % 255];
  const float iv   = s_alpha[invIdx];
  s_invA[t]  = iv;
  s_xorT[t]  = iv;             // reuse as reduction scratch (overwritten later)
  s_beta2[t + 16] = s_beta[t];
  __syncthreads();
  {
    const float ia0 = s_invA[0];
#pragma unroll
    for (int u = t; u < 768; u += 256) {
      const int uu = (u >= 255) ? (u - 255) : u;     // u in [0,509] -> mod 255
      s_G2[u] = (u < 510) ? s_invA[GFT.alog[uu]] : ia0;
    }
  }
#pragma unroll
  for (int stride = 128; stride > 0; stride >>= 1) {
    if (t < stride) s_xorT[t] += s_xorT[t + stride];
    __syncthreads();
  }
  if (t == 0) s_sum = s_xorT[0];
  __syncthreads();
  const float Ssum = s_sum;

  const int z0 = wave * 16 + m;   // N-tile #wave
  const int z1 = z0 + 128;        // N-tile #(wave+8), never 0

  // ---- Phase 3: GEMM1 — xorred = msbox x beta[x^z] ------------------------
  {
    // half==1 lanes read the +16-dword-shifted copy -> disjoint bank groups
    const char* __restrict__ betaP =
        half ? ((const char*)s_beta2 + 64) : (const char*)s_beta;
    const char* __restrict__ aB = (const char*)s_msbox + (m * 1024 + half * 8);
    const int zh0 = (z0 << 2) ^ (half << 3);
    const int zh1 = (z1 << 2) ^ (half << 3);

    v8f c0 = {0.f, 0.f, 0.f, 0.f, 0.f, 0.f, 0.f, 0.f};
    v8f c1 = {0.f, 0.f, 0.f, 0.f, 0.f, 0.f, 0.f, 0.f};
#pragma unroll
    for (int k4 = 0; k4 < 1024; k4 += 16) {        // k4 = k0*4 (byte offset)
      const v2f a = *(const v2f*)(aB + k4);        // ds_load_b64 (imm offset)
      const int p0 = k4 ^ zh0;
      const int p1 = k4 ^ zh1;
      v2f b0, b1;
      b0.x = *(const float*)(betaP + p0);
      b0.y = *(const float*)(betaP + (p0 ^ 4));
      b1.x = *(const float*)(betaP + p1);
      b1.y = *(const float*)(betaP + (p1 ^ 4));
      c0 = __builtin_amdgcn_wmma_f32_16x16x4_f32(false, a, false, b0, (short)0, c0, false, false);
      c1 = __builtin_amdgcn_wmma_f32_16x16x4_f32(false, a, false, b1, (short)0, c1, false, false);
    }
    __syncthreads();   // reduction scratch fully consumed before overwrite
    // stage xorred column-major [z][l] with pad 17 (bank-spread stores)
#pragma unroll
    for (int j = 0; j < 8; ++j) {
      const int row = j + 8 * half;
      s_xorT[z0 * 17 + row] = c0[j];
      s_xorT[z1 * 17 + row] = c1[j];
    }
  }
  __syncthreads();

  // ---- Phase 4: GEMM2 (exponent-reordered K) ------------------------------
  const char* __restrict__ xorTB = (const char*)s_xorT;
  const char* __restrict__ aoffB = (const char*)s_aoff + half * 8;
  const int m4 = m * 4;
  // per-lane loop-invariant B bases; z==0 lane -> invA0 region (index 509)
  const int Lz0 = (z0 == 0) ? 509 : (int)GFT.logt[z0];
  const int Lz1 = (int)GFT.logt[z1];               // z1 >= 128, never 0
  const float* __restrict__ g2p0 =
      (const float*)((const char*)s_G2 + Lz0 * 4 + half * 8);
  const float* __restrict__ g2p1 =
      (const float*)((const char*)s_G2 + Lz1 * 4 + half * 8);

  v8f d0 = {0.f, 0.f, 0.f, 0.f, 0.f, 0.f, 0.f, 0.f};
  v8f d1 = {0.f, 0.f, 0.f, 0.f, 0.f, 0.f, 0.f, 0.f};

#pragma unroll
  for (int jd = 0; jd < 252; jd += 4) {            // j = jd + 2*half + {0,1}
    const int2 ao = *(const int2*)(aoffB + jd * 4);  // uniform-ish b64, imm
    v2f a;
    a.x = *(const float*)(xorTB + (m4 + ao.x));    // xorred[m][alog[254-j]]
    a.y = *(const float*)(xorTB + (m4 + ao.y));
    v2f b0, b1;
    b0.x = g2p0[jd + 1];                           // -> ds_load_2addr_b32
    b0.y = g2p0[jd + 2];
    b1.x = g2p1[jd + 1];
    b1.y = g2p1[jd + 2];
    d0 = __builtin_amdgcn_wmma_f32_16x16x4_f32(false, a, false, b0, (short)0, d0, false, false);
    d1 = __builtin_amdgcn_wmma_f32_16x16x4_f32(false, a, false, b1, (short)0, d1, false, false);
  }
  {  // peeled last step: j = 252..255; j==255 is the y==0 fixup row
    const int2 ao = *(const int2*)(aoffB + 1008);
    v2f a;
    a.x = *(const float*)(xorTB + (m4 + ao.x));
    a.y = *(const float*)(xorTB + (m4 + ao.y));    // j=255 -> xorred[m][0]
    const float t0 = g2p0[253];
    const float t1 = g2p0[254];                    // junk for half==1 (safe)
    const float u0 = g2p1[253];
    const float u1 = g2p1[254];                    // junk for half==1 (safe)
    v2f b0, b1;
    b0.x = t0;
    b1.x = u0;
    b0.y = (half == 0) ? t1 : ((z0 == 0) ? Ssum : 0.f);
    b1.y = (half == 0) ? u1 : 0.f;
    d0 = __builtin_amdgcn_wmma_f32_16x16x4_f32(false, a, false, b0, (short)0, d0, false, false);
    d1 = __builtin_amdgcn_wmma_f32_16x16x4_f32(false, a, false, b1, (short)0, d1, false, false);
  }

  // ---- Phase 5: log(clip(out, 1e-12)) + store -----------------------------
  float* __restrict__ dst = out + (size_t)b * 16 * 256;
#pragma unroll
  for (int j = 0; j < 8; ++j) {
    const int row = j + 8 * half;
    dst[row * 256 + z0] = __logf(fmaxf(d0[j], 1e-12f));
    dst[row * 256 + z1] = __logf(fmaxf(d1[j], 1e-12f));
  }
}

extern "C" void kernel_launch(void* const* d_in, const int* in_sizes, int n_in,
                              void* d_out, int out_size, void* d_ws, size_t ws_size,
                              hipStream_t stream) {
  const float* logits = (const float*)d_in[0];
  float* out = (float*)d_out;
  const int B = in_sizes[0] / (18 * 256);   // 2048
  hipLaunchKernelGGL(ascadv2_head_kernel, dim3(B), dim3(256), 0, stream,
                     logits, out);
}